// gcn_layer_50706383897203
// MI455X (gfx1250) — compile-verified
//
#include <hip/hip_runtime.h>

typedef float v2f __attribute__((ext_vector_type(2)));
typedef float v8f __attribute__((ext_vector_type(8)));

#define INF  128
#define OUTF 128

// -----------------------------------------------------------------------------
// Phase 1: hidden[n][o] = sum_k x[n][k] * W[o][k] + b[o]
// One wave32 per 16-row node band; 8 accumulator tiles (the full 128 output
// features) so the A fragment is loaded once per k-step and reused by 8 WMMAs.
// V_WMMA_F32_16X16X4_F32 layouts:
//   A (16x4):  lane -> M = lane&15, VGPR j -> K = j + 2*(lane>>4)
//   B (4x16):  lane -> N = lane&15, VGPR j -> K = j + 2*(lane>>4)
//   C/D:       lane -> N = lane&15, VGPR v -> M = v + 8*(lane>>4)
// Bias depends only on column N -> accumulators are bias-initialized.
// -----------------------------------------------------------------------------
__global__ __launch_bounds__(256) void gemm_wmma_band(
    const float* __restrict__ x,
    const float* __restrict__ W,
    const float* __restrict__ b,
    float* __restrict__ hidden,
    int nBands)                      // N/16
{
    const int waveId = threadIdx.x >> 5;
    const int lane   = threadIdx.x & 31;
    const int band   = blockIdx.x * 8 + waveId;
    if (band >= nBands) return;

    const int nodeBase = band * 16;
    const int half = lane >> 4;
    const int l16  = lane & 15;

    v8f c[8];
#pragma unroll
    for (int t = 0; t < 8; ++t) {
        const float bias = b[t * 16 + l16];
#pragma unroll
        for (int i = 0; i < 8; ++i) c[t][i] = bias;
    }

    const float* xrow = x + (size_t)(nodeBase + l16) * INF + 2 * half;
    const float* wrow = W + (size_t)l16 * INF + 2 * half;  // tile t row = W[t*16+l16]

#pragma unroll 2
    for (int kb = 0; kb < INF; kb += 4) {
        v2f a = *(const v2f*)(xrow + kb);
#pragma unroll
        for (int t = 0; t < 8; ++t) {
            v2f bb = *(const v2f*)(wrow + (size_t)t * 16 * INF + kb);
            c[t] = __builtin_amdgcn_wmma_f32_16x16x4_f32(
                       false, a, false, bb, (short)0, c[t], false, false);
        }
    }

    float* outp = hidden + (size_t)(nodeBase + 8 * half) * OUTF + l16;
#pragma unroll
    for (int t = 0; t < 8; ++t) {
#pragma unroll
        for (int v = 0; v < 8; ++v) {
            outp[(size_t)v * OUTF + t * 16] = c[t][v];
        }
    }
}

// -----------------------------------------------------------------------------
// CSR build: histogram -> exclusive scan -> slot scatter.
// -----------------------------------------------------------------------------
__global__ __launch_bounds__(256) void zero_u32(unsigned* __restrict__ p, int n)
{
    int i = blockIdx.x * blockDim.x + threadIdx.x;
    if (i < n) p[i] = 0u;
}

__global__ __launch_bounds__(256) void hist_dst(
    const long long* __restrict__ edge_index, unsigned* __restrict__ counts, int E)
{
    int e = blockIdx.x * blockDim.x + threadIdx.x;
    if (e < E) atomicAdd(&counts[(int)edge_index[e]], 1u);
}

// Single-workgroup chunked Hillis-Steele scan over N bins (N=50000 -> 49 chunks).
__global__ __launch_bounds__(1024) void scan_offsets(
    const unsigned* __restrict__ counts,
    unsigned* __restrict__ offsets,
    unsigned* __restrict__ cursors, int n)
{
    __shared__ unsigned smem[1024];
    __shared__ unsigned running;
    if (threadIdx.x == 0) running = 0u;
    __syncthreads();

    for (int base = 0; base < n; base += 1024) {
        const int i = base + (int)threadIdx.x;
        const unsigned v = (i < n) ? counts[i] : 0u;
        smem[threadIdx.x] = v;
        __syncthreads();
#pragma unroll
        for (int off = 1; off < 1024; off <<= 1) {
            unsigned t = (threadIdx.x >= (unsigned)off) ? smem[threadIdx.x - off] : 0u;
            __syncthreads();
            smem[threadIdx.x] += t;
            __syncthreads();
        }
        const unsigned excl = smem[threadIdx.x] - v + running;
        if (i < n) { offsets[i] = excl; cursors[i] = excl; }
        __syncthreads();
        if (threadIdx.x == 1023) running += smem[1023];
        __syncthreads();
    }
    if (threadIdx.x == 0) offsets[n] = running;
}

__global__ __launch_bounds__(256) void scatter_csr(
    const long long* __restrict__ edge_index,
    const float*     __restrict__ edge_weight,
    unsigned* __restrict__ cursors,
    int*      __restrict__ csr_src,
    float*    __restrict__ csr_w, int E)
{
    int e = blockIdx.x * blockDim.x + threadIdx.x;
    if (e >= E) return;
    const int dst = (int)edge_index[e];
    const int src = (int)edge_index[(size_t)E + e];
    const unsigned pos = atomicAdd(&cursors[dst], 1u);
    csr_src[pos] = src;
    csr_w[pos]   = edge_weight[e];
}

// -----------------------------------------------------------------------------
// Phase 2 (CSR): one wave per destination node, float4 per lane.
// All accumulation in registers; single b128 store per lane. No f32 atomics.
// -----------------------------------------------------------------------------
__global__ __launch_bounds__(256) void gather_csr(
    const unsigned* __restrict__ offsets,
    const int*      __restrict__ csr_src,
    const float*    __restrict__ csr_w,
    const float*    __restrict__ hidden,
    float*          __restrict__ out, int N)
{
    const int waveId = threadIdx.x >> 5;
    const int lane   = threadIdx.x & 31;
    const int node   = blockIdx.x * 8 + waveId;
    if (node >= N) return;

    const unsigned s0 = offsets[node];
    const unsigned s1 = offsets[node + 1];
    const int col = lane * 4;

    float4 acc = make_float4(0.f, 0.f, 0.f, 0.f);
    for (unsigned j = s0; j < s1; ++j) {
        const int   src = csr_src[j];
        const float w   = csr_w[j];
        const float4 h = *(const float4*)(hidden + (size_t)src * OUTF + col);
        acc.x += w * h.x;
        acc.y += w * h.y;
        acc.z += w * h.z;
        acc.w += w * h.w;
    }
    *(float4*)(out + (size_t)node * OUTF + col) = acc;
}

// -----------------------------------------------------------------------------
// Fallback path (small workspace): direct atomic scatter.
// -----------------------------------------------------------------------------
__global__ __launch_bounds__(256) void zero_f32(float* __restrict__ p, int n)
{
    int i = blockIdx.x * blockDim.x + threadIdx.x;
    if (i < n) p[i] = 0.0f;
}

__global__ __launch_bounds__(256) void spmm_scatter(
    const long long* __restrict__ edge_index,
    const float*     __restrict__ edge_weight,
    const float*     __restrict__ hidden,
    float*           __restrict__ out, int E)
{
    const int t    = blockIdx.x * blockDim.x + threadIdx.x;
    const int e    = t >> 5;
    const int lane = t & 31;
    if (e >= E) return;
    const long long dst = edge_index[e];
    const long long src = edge_index[(size_t)E + e];
    const float     w   = edge_weight[e];
    const float4 hv = *(const float4*)(hidden + (size_t)src * OUTF + lane * 4);
    float* o = out + (size_t)dst * OUTF + lane * 4;
    atomicAdd(o + 0, w * hv.x);
    atomicAdd(o + 1, w * hv.y);
    atomicAdd(o + 2, w * hv.z);
    atomicAdd(o + 3, w * hv.w);
}

// -----------------------------------------------------------------------------
// Inputs: 0:x [N,128] f32, 1:edge_index [2,E] i64, 2:edge_weight [E] f32,
//         3:W [128,128] f32, 4:b [128] f32.   d_out: [N,128] f32.
// Workspace layout: hidden | offsets(N+1) | cursors(N) | csr_src(E) | csr_w(E)
// -----------------------------------------------------------------------------
extern "C" void kernel_launch(void* const* d_in, const int* in_sizes, int n_in,
                              void* d_out, int out_size, void* d_ws, size_t ws_size,
                              hipStream_t stream)
{
    const float*     x  = (const float*)d_in[0];
    const long long* ei = (const long long*)d_in[1];
    const float*     ew = (const float*)d_in[2];
    const float*     W  = (const float*)d_in[3];
    const float*     b  = (const float*)d_in[4];
    float* out = (float*)d_out;

    const int N = in_sizes[0] / INF;   // 50000
    const int E = in_sizes[2];         // 800000

    // workspace carve-up
    float*    hidden  = (float*)d_ws;
    size_t    off     = (size_t)N * OUTF;              // floats
    unsigned* offsets = (unsigned*)(hidden + off);     // N+1
    unsigned* cursors = offsets + (N + 1);             // N
    int*      csr_src = (int*)(cursors + N);           // E
    float*    csr_w   = (float*)(csr_src + E);         // E
    const size_t need = ((size_t)N * OUTF + (size_t)(2 * N + 1) + 2 * (size_t)E) * 4;

    // Phase 1: GEMM via WMMA (one wave per 16-row band, 8 tiles per wave)
    {
        const int nBands = N / 16;                 // 3125
        const int blocks = (nBands + 7) / 8;       // 391
        gemm_wmma_band<<<blocks, 256, 0, stream>>>(x, W, b, hidden, nBands);
    }

    if (ws_size >= need) {
        // ---- CSR path: build adjacency then register-accumulated gather ----
        zero_u32<<<(N + 1 + 255) / 256, 256, 0, stream>>>(offsets, N + 1);
        hist_dst<<<(E + 255) / 256, 256, 0, stream>>>(ei, offsets, E);
        // note: histogram accumulates into `offsets`; scan reads it as counts
        scan_offsets<<<1, 1024, 0, stream>>>(offsets, offsets, cursors, N);
        scatter_csr<<<(E + 255) / 256, 256, 0, stream>>>(ei, ew, cursors, csr_src, csr_w, E);
        gather_csr<<<(N + 7) / 8, 256, 0, stream>>>(offsets, csr_src, csr_w, hidden, out, N);
    } else {
        // ---- fallback: direct atomic scatter ----
        const int n = N * OUTF;
        zero_f32<<<(n + 255) / 256, 256, 0, stream>>>(out, n);
        const long long threads = (long long)E * 32;
        spmm_scatter<<<(int)((threads + 255) / 256), 256, 0, stream>>>(ei, ew, hidden, out, E);
    }
}